// ODNet_att_21431886807157
// MI455X (gfx1250) — compile-verified
//
#include <hip/hip_runtime.h>
#include <hip/hip_bf16.h>

// ---------------- problem constants ----------------
#define BB    64                    // batch of graphs
#define NN    512                   // nodes per graph
#define BN_T  32768                 // BB*NN
#define DIN_T 256
#define U_T   128
#define D4_T  512                   // 4*U
#define E_T   262144                // BN*DEG
#define GEFF  768                   // effective gconv input: 3 hops x 256 (h-part is zero)

typedef float v2f __attribute__((ext_vector_type(2)));
typedef float v8f __attribute__((ext_vector_type(8)));
typedef unsigned int u32x4 __attribute__((ext_vector_type(4)));
typedef int i32x4 __attribute__((ext_vector_type(4)));
typedef int i32x8 __attribute__((ext_vector_type(8)));

// Tensor Data Mover availability (this toolchain: 6-arg builtin, clang-23 style)
#ifdef __has_builtin
#  if __has_builtin(__builtin_amdgcn_tensor_load_to_lds) && __has_builtin(__builtin_amdgcn_s_wait_tensorcnt)
#    define USE_TDM 1
#  else
#    define USE_TDM 0
#  endif
#else
#  define USE_TDM 0
#endif

// =====================================================================
// Generic batched WMMA-f32 GEMM:  C = alpha * A@B(+T) + beta*C + bias
// A: MxK row-major (lda), B: KxN row-major (ldb) or NxK if transB.
// Block: 256 thr = 8 waves (2 M x 4 N), block tile 64x64, K chunk 32.
// Each wave owns two stacked 16x16 accumulators -> 16 WMMA per chunk.
// A (and B when !transB) panels are DMA'd into LDS by the TDM with the
// bank-conflict padding generated by the descriptor's pad fields.
// Requires M%64==0, N%64==0, K%32==0 (true for every call here).
// =====================================================================
#define TM 64
#define TN 64
#define TK 32

__global__ __launch_bounds__(256)
void gemm_wmma_f32(const float* __restrict__ A, long long lda, long long sAb,
                   const float* __restrict__ B, long long ldb, long long sBb, int transB,
                   float* __restrict__ C, long long ldc, long long sCb,
                   int M, int N, int K, float alpha, float beta,
                   const float* __restrict__ bias)
{
    __shared__ float As[TM][TK + 1];   // 64 x 33: +1 pad == TDM pad(interval 32, 1 dword)
    __shared__ float Bs[TK][TN + 1];   // 32 x 65: +1 pad == TDM pad(interval 64, 1 dword)

    const int z = blockIdx.z;
    A += (long long)z * sAb;
    B += (long long)z * sBb;
    C += (long long)z * sCb;

    const int m0   = blockIdx.y * TM;
    const int n0   = blockIdx.x * TN;
    const int tid  = threadIdx.x;
    const int lane = tid & 31;
    const int wid  = tid >> 5;
    const int wm   = wid >> 2;          // 0..1
    const int wn   = wid & 3;           // 0..3
    const int half = lane >> 4;         // 0: lanes 0-15, 1: lanes 16-31
    const int l16  = lane & 15;

    v8f acc0 = {}, acc1 = {};

#if USE_TDM
    // Flat-LDS generic pointers: low 32 bits == LDS byte offset (ISA 10.2).
    const unsigned ldsA = (unsigned)(unsigned long long)(void*)&As[0][0];
    const unsigned ldsB = (unsigned)(unsigned long long)(void*)&Bs[0][0];
    // D# group1 constants: data_size=4B, pad_enable, pad_interval code, pad_amount=1 dword.
    const int g1a_flags = (int)((2u << 16) | (1u << 20) | (4u << 22));  // interval 32 dwords
    const int g1b_flags = (int)((2u << 16) | (1u << 20) | (5u << 22));  // interval 64 dwords
    const i32x4 gz4 = {0, 0, 0, 0};
    const i32x8 gz8 = {0, 0, 0, 0, 0, 0, 0, 0};
#endif

    for (int k0 = 0; k0 < K; k0 += TK) {
#if USE_TDM
        if (wid == 0) {
            // ---- A tile via TDM: tile 32(k) x 64(m rows), row stride lda ----
            {
                unsigned long long ga =
                    (unsigned long long)(A + (long long)m0 * lda + k0);
                u32x4 g0 = { 1u, ldsA, (unsigned)ga,
                             (unsigned)((ga >> 32) & 0x01FFFFFFu) | (2u << 30) };
                i32x8 g1 = { g1a_flags,
                             0,                                   // atomic_barrier_addr
                             0x4000,                              // tensor_dim0 hi (1<<30)
                             (int)(0x4000u | (32u << 16)),        // tensor_dim1 hi | tile_dim0=32
                             64,                                  // tile_dim1=64, tile_dim2=0
                             (int)(unsigned)lda,                  // dim0 stride lo
                             (int)(unsigned)((unsigned long long)lda >> 32),
                             0 };
                __builtin_amdgcn_tensor_load_to_lds(g0, g1, gz4, gz4, gz8, 0);
            }
            // ---- B tile via TDM (row-major case): 64(n) x 32(k rows) ----
            if (!transB) {
                unsigned long long gb =
                    (unsigned long long)(B + (long long)k0 * ldb + n0);
                u32x4 g0 = { 1u, ldsB, (unsigned)gb,
                             (unsigned)((gb >> 32) & 0x01FFFFFFu) | (2u << 30) };
                i32x8 g1 = { g1b_flags,
                             0,
                             0x4000,
                             (int)(0x4000u | (64u << 16)),        // tile_dim0=64
                             32,                                  // tile_dim1=32
                             (int)(unsigned)ldb,
                             (int)(unsigned)((unsigned long long)ldb >> 32),
                             0 };
                __builtin_amdgcn_tensor_load_to_lds(g0, g1, gz4, gz4, gz8, 0);
            }
            __builtin_amdgcn_s_wait_tensorcnt(0);
        }
        if (transB) {
            const int kk = tid >> 3;            // 0..31
            const int cb = (tid & 7) << 3;      // 0,8,...,56
            #pragma unroll
            for (int i = 0; i < 8; ++i)
                Bs[kk][cb + i] = B[(long long)(n0 + cb + i) * ldb + (k0 + kk)];
        }
#else
        // ---- fallback: vector loads + LDS stores ----
        {
            const int r  = tid >> 2;            // 0..63
            const int cb = (tid & 3) << 3;      // 0,8,16,24
            const float* ap = A + (long long)(m0 + r) * lda + (k0 + cb);
            #pragma unroll
            for (int i = 0; i < 8; ++i) As[r][cb + i] = ap[i];
            if (k0 + TK < K) __builtin_prefetch(ap + TK, 0, 1);
        }
        {
            const int kk = tid >> 3;            // 0..31
            const int cb = (tid & 7) << 3;      // 0,8,...,56
            if (!transB) {
                const float* bp = B + (long long)(k0 + kk) * ldb + (n0 + cb);
                #pragma unroll
                for (int i = 0; i < 8; ++i) Bs[kk][cb + i] = bp[i];
                if (k0 + TK < K) __builtin_prefetch(bp + (long long)TK * ldb, 0, 1);
            } else {
                #pragma unroll
                for (int i = 0; i < 8; ++i)
                    Bs[kk][cb + i] = B[(long long)(n0 + cb + i) * ldb + (k0 + kk)];
            }
        }
#endif
        __syncthreads();

        // ---- 16 x V_WMMA_F32_16X16X4_F32 per wave per chunk ----
        const int mr0  = wm * 32 + l16;
        const int mr1  = mr0 + 16;
        const int ncol = wn * 16 + l16;
        #pragma unroll
        for (int kk = 0; kk < TK; kk += 4) {
            // A frag: VGPR0 = K0 (lanes<16) / K2 (lanes>=16); VGPR1 = K1/K3
            v2f a0, a1, b;
            a0.x = As[mr0][kk + half * 2 + 0];
            a0.y = As[mr0][kk + half * 2 + 1];
            a1.x = As[mr1][kk + half * 2 + 0];
            a1.y = As[mr1][kk + half * 2 + 1];
            b.x  = Bs[kk + half * 2 + 0][ncol];
            b.y  = Bs[kk + half * 2 + 1][ncol];
            acc0 = __builtin_amdgcn_wmma_f32_16x16x4_f32(
                       false, a0, false, b, (short)0, acc0, false, false);
            acc1 = __builtin_amdgcn_wmma_f32_16x16x4_f32(
                       false, a1, false, b, (short)0, acc1, false, false);
        }
        __syncthreads();
    }

    // ---- epilogue: VGPR v -> M = v (lanes 0-15) / v+8 (lanes 16-31) ----
    const int gm0  = m0 + wm * 32 + (half ? 8 : 0);
    const int gn   = n0 + wn * 16 + l16;
    const float bval = bias ? bias[gn] : 0.0f;
    #pragma unroll
    for (int v = 0; v < 8; ++v) {
        const long long ci0 = (long long)(gm0 + v) * ldc + gn;
        const long long ci1 = (long long)(gm0 + 16 + v) * ldc + gn;
        float v0 = alpha * acc0[v] + bval;
        float v1 = alpha * acc1[v] + bval;
        if (beta != 0.0f) { v0 += beta * C[ci0]; v1 += beta * C[ci1]; }
        C[ci0] = v0;
        C[ci1] = v1;
    }
}

// =====================================================================
// Graph kernels
// =====================================================================
__global__ void k_deg_scatter(float* __restrict__ deg, const int* __restrict__ dst,
                              const float* __restrict__ ew)
{
    const int e = blockIdx.x * 256 + threadIdx.x;       // grid = E/256
    unsafeAtomicAdd(&deg[dst[e]], ew[e]);
}

// out[dst] += ew * in[src], 4 features per thread (grid = E*64/256)
__global__ void k_spmm_scatter(float* __restrict__ out, const float* __restrict__ in,
                               const int* __restrict__ src, const int* __restrict__ dst,
                               const float* __restrict__ ew)
{
    const long long idx = (long long)blockIdx.x * 256 + threadIdx.x;
    const int e = (int)(idx >> 6);
    const int f = (int)(idx & 63) << 2;
    const float w = ew[e];
    const float4 v = *(const float4*)(in + (long long)src[e] * DIN_T + f);
    float* op = out + (long long)dst[e] * DIN_T + f;
    unsafeAtomicAdd(op + 0, w * v.x);
    unsafeAtomicAdd(op + 1, w * v.y);
    unsafeAtomicAdd(op + 2, w * v.z);
    unsafeAtomicAdd(op + 3, w * v.w);
}

__global__ void k_normalize(float* __restrict__ X, const float* __restrict__ deg)
{
    const long long idx = (long long)blockIdx.x * 256 + threadIdx.x;  // BN*64
    const int i = (int)(idx >> 6);
    const int f = (int)(idx & 63) << 2;
    const float inv = 1.0f / fmaxf(deg[i], 1.0f);
    float4* p = (float4*)(X + (long long)i * DIN_T + f);
    float4 v = *p;
    v.x *= inv; v.y *= inv; v.z *= inv; v.w *= inv;
    *p = v;
}

// Gather effective branch weight (768x256): cols 0..127 = u-logit, 128..255 = c-logit.
__global__ void k_build_weff(float* __restrict__ Weff, float* __restrict__ beff,
                             const float* __restrict__ W_ru, const float* __restrict__ b_ru,
                             const float* __restrict__ W_c,  const float* __restrict__ b_c)
{
    const int idx = blockIdx.x * 256 + threadIdx.x;     // 768*256
    const int k = idx >> 8;                             // effective row
    const int j = idx & 255;                            // output column
    const int t = k >> 8;                               // hop
    const int d = k & 255;                              // input feature
    const int row = t * (DIN_T + U_T) + d;              // row in full 1152-row weight
    Weff[idx] = (j < 128) ? W_ru[(long long)row * 256 + 128 + j]
                          : W_c [(long long)row * 128 + (j - 128)];
    if (k == 0) beff[j] = (j < 128) ? b_ru[128 + j] : b_c[j - 128];
}

// out = (1 - sigmoid(u)) * tanh(c)
__global__ void k_branch_act(float* __restrict__ out, const float* __restrict__ logits)
{
    const long long idx = (long long)blockIdx.x * 256 + threadIdx.x;  // BN*128
    const int m = (int)(idx >> 7);
    const int j = (int)(idx & 127);
    const float u = logits[(long long)m * 256 + j];
    const float c = logits[(long long)m * 256 + 128 + j];
    const float su = 1.0f / (1.0f + __expf(-u));
    out[idx] = (1.0f - su) * tanhf(c);
}

// Bf[c*32768 + b*512 + n] = fusion[b,c,n]  (raw .view of concat([hs,hl]))
__global__ void k_build_bf(float* __restrict__ Bf, const float* __restrict__ hs,
                           const float* __restrict__ hl)
{
    const long long idx = (long long)blockIdx.x * 256 + threadIdx.x;  // 256*32768
    const int c   = (int)(idx >> 15);
    const int rem = (int)(idx & 32767);
    const int b   = rem >> 9;
    const int n   = rem & 511;
    const int f   = c * NN + n;
    const int no  = f >> 8;         // original node
    const int co  = f & 255;        // original channel in concat([hs,hl])
    const float* s = (co < 128) ? hs : hl;
    Bf[idx] = s[((long long)b * NN + no) * U_T + (co & 127)];
}

// uh_mat[o, b*512+n] += b_hd[o] + pe(b, o)
__global__ void k_uh_epilogue(float* __restrict__ uh, const float* __restrict__ b_hd)
{
    const long long idx = (long long)blockIdx.x * 256 + threadIdx.x;  // 512*32768
    const int o   = (int)(idx >> 15);
    const int col = (int)(idx & 32767);
    const int b   = col >> 9;
    const int i2  = o & ~1;
    const float freq = __expf(-(float)i2 * 0.017988946039015984f);   // ln(10000)/512
    const float ang  = (float)b * freq;
    uh[idx] += b_hd[o] + ((o & 1) ? __cosf(ang) : __sinf(ang));
}

// x_att[(s*64+b)*512 + c] = uh_mat[c*32768 + b*512 + s]
__global__ void k_build_xatt(float* __restrict__ xatt, const float* __restrict__ uh)
{
    const long long idx = (long long)blockIdx.x * 256 + threadIdx.x;  // 32768*512
    const int row = (int)(idx >> 9);
    const int c   = (int)(idx & 511);
    const int s   = row >> 6;
    const int b   = row & 63;
    xatt[idx] = uh[(long long)c * 32768 + b * NN + s];
}

// Softmax over last dim (rowlen=512), one row per block.
__global__ void k_softmax(float* __restrict__ S, int rowlen)
{
    __shared__ float red[256];
    float* row = S + (long long)blockIdx.x * rowlen;
    const int tid = threadIdx.x;
    float m = -3.4e38f;
    for (int i = tid; i < rowlen; i += 256) m = fmaxf(m, row[i]);
    red[tid] = m; __syncthreads();
    for (int s = 128; s > 0; s >>= 1) { if (tid < s) red[tid] = fmaxf(red[tid], red[tid + s]); __syncthreads(); }
    m = red[0]; __syncthreads();
    float sum = 0.0f;
    for (int i = tid; i < rowlen; i += 256) { const float e = __expf(row[i] - m); row[i] = e; sum += e; }
    red[tid] = sum; __syncthreads();
    for (int s = 128; s > 0; s >>= 1) { if (tid < s) red[tid] += red[tid + s]; __syncthreads(); }
    const float inv = 1.0f / red[0];
    for (int i = tid; i < rowlen; i += 256) row[i] *= inv;
}

// Aperm[(b*512+n)*512 + c] = attn_out[(n*64+b)*512 + c]
__global__ void k_permute_gate(float* __restrict__ Ap, const float* __restrict__ ao)
{
    const long long idx = (long long)blockIdx.x * 256 + threadIdx.x;  // 32768*512
    const int row = (int)(idx >> 9);
    const int c   = (int)(idx & 511);
    const int b   = row >> 9;
    const int n   = row & 511;
    Ap[idx] = ao[((long long)n * 64 + b) * 512 + c];
}

// =====================================================================
// Host-side orchestration
// =====================================================================
static inline void launch_gemm(hipStream_t st,
                               const float* A, long long lda, long long sA,
                               const float* B, long long ldb, long long sB, int transB,
                               float* C, long long ldc, long long sC,
                               int M, int N, int K, float alpha, float beta,
                               const float* bias, int batch)
{
    dim3 g(N / TN, M / TM, batch), b(256);
    hipLaunchKernelGGL(gemm_wmma_f32, g, b, 0, st,
                       A, lda, sA, B, ldb, sB, transB, C, ldc, sC,
                       M, N, K, alpha, beta, bias);
}

extern "C" void kernel_launch(void* const* d_in, const int* in_sizes, int n_in,
                              void* d_out, int out_size, void* d_ws, size_t ws_size,
                              hipStream_t stream)
{
    const float* xin[3]  = {(const float*)d_in[0], (const float*)d_in[1], (const float*)d_in[2]};
    const int*   eidx    = (const int*)d_in[3];
    const float* ew      = (const float*)d_in[4];
    const float* W_ru[3] = {(const float*)d_in[5],  (const float*)d_in[9],  (const float*)d_in[13]};
    const float* b_ru[3] = {(const float*)d_in[6],  (const float*)d_in[10], (const float*)d_in[14]};
    const float* W_c[3]  = {(const float*)d_in[7],  (const float*)d_in[11], (const float*)d_in[15]};
    const float* b_c[3]  = {(const float*)d_in[8],  (const float*)d_in[12], (const float*)d_in[16]};
    const float* W_hd   = (const float*)d_in[17];
    const float* b_hd   = (const float*)d_in[18];
    const float* W_in   = (const float*)d_in[19];
    const float* b_in   = (const float*)d_in[20];
    const float* W_out  = (const float*)d_in[21];
    const float* b_out  = (const float*)d_in[22];
    const float* W_gate = (const float*)d_in[23];
    const float* b_gate = (const float*)d_in[24];

    const int* srcIdx = eidx;
    const int* dstIdx = eidx + E_T;
    float* out = (float*)d_out;

    // ---- workspace layout (64MB slots, liveness-based reuse) ----
    char* ws = (char*)d_ws;
    float* deg  = (float*)(ws);                               // 128KB
    float* Weff = (float*)(ws + (512ull << 10));              // 768KB
    float* beff = (float*)(ws + (512ull << 10) + GEFF * 256 * 4);
    const size_t HDR = 2ull << 20, SLOT = 64ull << 20;
    float* S0 = (float*)(ws + HDR + 0 * SLOT);
    float* S1 = (float*)(ws + HDR + 1 * SLOT);
    float* S2 = (float*)(ws + HDR + 2 * SLOT);
    float* S3 = (float*)(ws + HDR + 3 * SLOT);
    float* S4 = (float*)(ws + HDR + 4 * SLOT);
    float* S5 = (float*)(ws + HDR + 5 * SLOT);
    float* S6 = (float*)(ws + HDR + 6 * SLOT);

    float* X1     = S0;                 // 32MB
    float* X2     = S0 + (1u << 23);    // 32MB
    float* logits = S1;                 // 32MB
    float* Bf     = S1 + (1u << 23);    // 32MB
    float* scores = S0;                 // 128MB, spans S0+S1 (X1/X2/logits/Bf dead)
    float* uh     = S2;                 // 64MB; later attn_out
    float* xatt   = S3;                 // 64MB; later ctx
    float* q      = S4;                 // 64MB; later Aperm
    float* k      = S5;
    float* v      = S6;
    float* ctx    = S3;
    float* aout   = S2;
    float* Aperm  = S4;

    // ---- degree ----
    (void)hipMemsetAsync(deg, 0, BN_T * sizeof(float), stream);
    hipLaunchKernelGGL(k_deg_scatter, dim3(E_T / 256), dim3(256), 0, stream, deg, dstIdx, ew);

    // ---- three GRU branches: hf -> out[0], hl -> out[1], hs -> out[2] ----
    for (int br = 0; br < 3; ++br) {
        (void)hipMemsetAsync(X1, 0, (size_t)BN_T * DIN_T * 4, stream);
        hipLaunchKernelGGL(k_spmm_scatter, dim3(E_T * 64 / 256), dim3(256), 0, stream,
                           X1, xin[br], srcIdx, dstIdx, ew);
        hipLaunchKernelGGL(k_normalize, dim3(BN_T * 64 / 256), dim3(256), 0, stream, X1, deg);
        (void)hipMemsetAsync(X2, 0, (size_t)BN_T * DIN_T * 4, stream);
        hipLaunchKernelGGL(k_spmm_scatter, dim3(E_T * 64 / 256), dim3(256), 0, stream,
                           X2, X1, srcIdx, dstIdx, ew);
        hipLaunchKernelGGL(k_normalize, dim3(BN_T * 64 / 256), dim3(256), 0, stream, X2, deg);

        hipLaunchKernelGGL(k_build_weff, dim3(GEFF * 256 / 256), dim3(256), 0, stream,
                           Weff, beff, W_ru[br], b_ru[br], W_c[br], b_c[br]);

        // logits = [x|X1|X2] @ Weff + beff   (3 accumulating K=256 panels)
        launch_gemm(stream, xin[br], 256, 0, Weff,            256, 0, 0,
                    logits, 256, 0, BN_T, 256, 256, 1.0f, 0.0f, beff, 1);
        launch_gemm(stream, X1,      256, 0, Weff + 256 * 256, 256, 0, 0,
                    logits, 256, 0, BN_T, 256, 256, 1.0f, 1.0f, nullptr, 1);
        launch_gemm(stream, X2,      256, 0, Weff + 512 * 256, 256, 0, 0,
                    logits, 256, 0, BN_T, 256, 256, 1.0f, 1.0f, nullptr, 1);

        hipLaunchKernelGGL(k_branch_act, dim3(BN_T * U_T / 256), dim3(256), 0, stream,
                           out + (size_t)br * BN_T * U_T, logits);
    }

    // ---- fusion GEMM: uh_mat (512 x 32768) = W_hd @ Bf ----
    hipLaunchKernelGGL(k_build_bf, dim3(256 * 32768 / 256), dim3(256), 0, stream,
                       Bf, out + 2ull * BN_T * U_T /*hs*/, out + 1ull * BN_T * U_T /*hl*/);
    launch_gemm(stream, W_hd, 256, 0, Bf, 32768, 0, 0,
                uh, 32768, 0, 512, 32768, 256, 1.0f, 0.0f, nullptr, 1);
    hipLaunchKernelGGL(k_uh_epilogue, dim3(512 * 32768 / 256), dim3(256), 0, stream, uh, b_hd);

    // ---- QKV: x_att (32768 x 512), q/k/v = x_att @ W_in[part].T + b_in ----
    hipLaunchKernelGGL(k_build_xatt, dim3(32768 * 512 / 256), dim3(256), 0, stream, xatt, uh);
    launch_gemm(stream, xatt, 512, 0, W_in,                 512, 0, 1,
                q, 512, 0, BN_T, 512, 512, 1.0f, 0.0f, b_in, 1);
    launch_gemm(stream, xatt, 512, 0, W_in + 512ull * 512,  512, 0, 1,
                k, 512, 0, BN_T, 512, 512, 1.0f, 0.0f, b_in + 512, 1);
    launch_gemm(stream, xatt, 512, 0, W_in + 1024ull * 512, 512, 0, 1,
                v, 512, 0, BN_T, 512, 512, 1.0f, 0.0f, b_in + 1024, 1);

    // ---- scores (batch z = b*2+h): A = q[., z*256+d] lda=32768, B = k^T ----
    launch_gemm(stream, q, 32768, 256, k, 32768, 256, 1,
                scores, 512, 512ll * 512, 512, 512, 256, 0.0625f, 0.0f, nullptr, 128);
    hipLaunchKernelGGL(k_softmax, dim3(128 * 512), dim3(256), 0, stream, scores, 512);

    // ---- ctx = attn @ V, lands directly as (s,b,h,d) row-major 32768x512 ----
    launch_gemm(stream, scores, 512, 512ll * 512, v, 32768, 256, 0,
                ctx, 32768, 256, 512, 256, 512, 1.0f, 0.0f, nullptr, 128);

    // ---- attn_out = ctx @ W_out.T + b_out ----
    launch_gemm(stream, ctx, 512, 0, W_out, 512, 0, 1,
                aout, 512, 0, BN_T, 512, 512, 1.0f, 0.0f, b_out, 1);

    // ---- gate: out[3] = Aperm @ W_gate.T + b_gate ----
    hipLaunchKernelGGL(k_permute_gate, dim3(32768 * 512 / 256), dim3(256), 0, stream, Aperm, aout);
    launch_gemm(stream, Aperm, 512, 0, W_gate, 512, 0, 1,
                out + 3ull * BN_T * U_T, 128, 0, BN_T, 128, 512, 1.0f, 0.0f, b_gate, 1);
}